// Electrostatics_13005160972686
// MI455X (gfx1250) — compile-verified
//
#include <hip/hip_runtime.h>
#include <hip/hip_bf16.h>

#define N_MOLS        4096
#define ATOMS_PER_MOL 60
#define FEAT_DIM      128
#define N_ATOMS       (N_MOLS * ATOMS_PER_MOL)

#define KE_KCAL   332.0637198f
#define R_ON      1.25f
#define R_OFF     3.75f
#define BOHR2     (0.529177f * 0.529177f)
#define EPS_F     1e-15f

typedef __attribute__((ext_vector_type(2))) float v2f;
typedef __attribute__((ext_vector_type(8))) float v8f;

// ---------------------------------------------------------------------------
// Kernel 1: charge[a] = dot(f[a,:], W) + z_embed[z[a]]  via V_WMMA_F32_16X16X4_F32
// One wave (32 lanes) computes a 16-row tile. B = W broadcast to all 16 cols,
// so every column of D holds the same per-row dot product.
// A-fragment layout (ISA 16x4 f32): lanes 0-15 -> rows 0-15 with K=0,1 in
// v0,v1; lanes 16-31 -> rows 0-15 with K=2,3.
// ---------------------------------------------------------------------------
__global__ void charge_gemv_wmma(const float* __restrict__ f,
                                 const int*   __restrict__ z,
                                 const float* __restrict__ z_embed,
                                 const float* __restrict__ W,
                                 float*       __restrict__ charge_out)
{
    const int lane = threadIdx.x & 31;
    const int wave = threadIdx.x >> 5;
    const int tile = blockIdx.x * (blockDim.x >> 5) + wave;
    const int half = lane >> 4;     // 0: K=0,1   1: K=2,3
    const int mrow = lane & 15;

    const long base = (long)tile * 16;
    const float* __restrict__ arow = f + (base + mrow) * FEAT_DIM + 2 * half;
    const float* __restrict__ wptr = W + 2 * half;

    v8f c = {};
#pragma unroll
    for (int kc = 0; kc < FEAT_DIM / 4; ++kc) {
        v2f a = *(const v2f*)(arow + 4 * kc);   // f[row][4kc+2h], f[row][4kc+2h+1]
        v2f b = *(const v2f*)(wptr + 4 * kc);   // W  [4kc+2h],    W  [4kc+2h+1] (bcast cols)
        // (neg_a, A, neg_b, B, c_mod, C, reuse_a, reuse_b)
        c = __builtin_amdgcn_wmma_f32_16x16x4_f32(false, a, false, b,
                                                  (short)0, c, false, false);
    }

    // D layout: VGPR v, lanes 0-15 -> M=v ; lanes 16-31 -> M=8+v. Columns all
    // identical, so lanes 0 and 16 each emit 8 rows.
    if (mrow == 0) {
#pragma unroll
        for (int v = 0; v < 8; ++v) {
            long row = base + v + 8 * half;
            charge_out[row] = c[v] + z_embed[z[row]];
        }
    }
}

// ---------------------------------------------------------------------------
// Kernel 2: per-molecule neutralization. q = charge + (Q_tot - sum)/60.
// charge lives in the q-region of d_out and is overwritten in place.
// ---------------------------------------------------------------------------
__global__ void mol_correct(float*       __restrict__ q,           // in: charge, out: q
                            const float* __restrict__ total_charge)
{
    __shared__ float s[64];
    const int m = blockIdx.x;
    const int t = threadIdx.x;

    float c = 0.0f;
    if (t < ATOMS_PER_MOL) c = q[(long)m * ATOMS_PER_MOL + t];
    s[t] = c;
    __syncthreads();
#pragma unroll
    for (int off = 32; off > 0; off >>= 1) {
        if (t < off) s[t] += s[t + off];
        __syncthreads();
    }
    const float corr = (total_charge[m] - s[0]) / (float)ATOMS_PER_MOL;
    if (t < ATOMS_PER_MOL) q[(long)m * ATOMS_PER_MOL + t] = c + corr;
}

// ---------------------------------------------------------------------------
// Kernel 3: pairwise switched-Coulomb energy per molecule. One block per
// molecule; 60 atoms staged in LDS; all 60*59 ordered pairs summed.
// ---------------------------------------------------------------------------
__global__ void pair_energy(const float* __restrict__ xyz,
                            const float* __restrict__ q,
                            float*       __restrict__ energy)
{
    __shared__ float sx[ATOMS_PER_MOL], sy[ATOMS_PER_MOL],
                     sz[ATOMS_PER_MOL], sq[ATOMS_PER_MOL];
    __shared__ float red[4];

    const int m = blockIdx.x;
    const int t = threadIdx.x;

    if (t < ATOMS_PER_MOL) {
        const long a = (long)m * ATOMS_PER_MOL + t;
        sx[t] = xyz[a * 3 + 0];
        sy[t] = xyz[a * 3 + 1];
        sz[t] = xyz[a * 3 + 2];
        sq[t] = q[a];
    }
    __syncthreads();

    const float inv_range = 1.0f / (R_OFF - R_ON);
    float acc = 0.0f;

    for (int cell = t; cell < ATOMS_PER_MOL * ATOMS_PER_MOL; cell += 128) {
        const int i = cell / ATOMS_PER_MOL;
        const int j = cell - i * ATOMS_PER_MOL;
        if (i == j) continue;

        const float dx = sx[i] - sx[j];
        const float dy = sy[i] - sy[j];
        const float dz = sz[i] - sz[j];
        const float r2 = fmaf(dx, dx, EPS_F) + fmaf(dy, dy, EPS_F) + fmaf(dz, dz, EPS_F);
        const float r  = sqrtf(r2);

        // f_switch(r), faithful to the reference branch structure
        const float x = (r - R_ON) * inv_range;
        const float y = 1.0f - x;
        const bool  interior = (x > 0.0f) && (y > 0.0f);
        const float denom    = interior ? (x * y) : 1.0f;
        const float exp_arg  = (x - y) / denom;
        const float e        = fminf(exp_arg, 34.0f);
        const float sig      = 1.0f / (1.0f + __expf(e));
        const float int_val  = (exp_arg >= 34.0f) ? 0.0f : sig;
        const float fs = ((x <= 0.0f) && (y > 0.0f)) ? 1.0f
                         : (interior ? int_val : 0.0f);

        const float arg0 = fs / sqrtf(fmaf(r, r, BOHR2));
        const float arg1 = (1.0f - fs) / r;
        acc += KE_KCAL * sq[i] * sq[j] * (arg0 + arg1);
    }

    // wave32 reduction, then cross-wave via LDS
#pragma unroll
    for (int off = 16; off > 0; off >>= 1)
        acc += __shfl_down(acc, off, 32);
    if ((t & 31) == 0) red[t >> 5] = acc;
    __syncthreads();
    if (t == 0) {
        float etot = red[0] + red[1] + red[2] + red[3];
        energy[m] = etot;
    }
}

// ---------------------------------------------------------------------------
extern "C" void kernel_launch(void* const* d_in, const int* in_sizes, int n_in,
                              void* d_out, int out_size, void* d_ws, size_t ws_size,
                              hipStream_t stream)
{
    (void)in_sizes; (void)n_in; (void)out_size; (void)d_ws; (void)ws_size;

    const float* f            = (const float*)d_in[0];
    const int*   z            = (const int*)  d_in[1];
    const float* xyz          = (const float*)d_in[2];
    const float* total_charge = (const float*)d_in[3];
    // d_in[4] num_atoms: uniformly 60  (unused)
    // d_in[5] mol_nbrs : all intra-molecule ordered pairs (regenerated implicitly)
    const float* W            = (const float*)d_in[6];
    const float* z_embed      = (const float*)d_in[7];

    float* energy = (float*)d_out;            // [N_MOLS]
    float* q      = (float*)d_out + N_MOLS;   // [N_ATOMS]; first holds raw charge

    // K1: 15360 tiles of 16 atoms, 8 waves (256 thr) per block -> 1920 blocks
    charge_gemv_wmma<<<(N_ATOMS / 16) / 8, 256, 0, stream>>>(f, z, z_embed, W, q);

    // K2: per-molecule charge correction (in place in the q region)
    mol_correct<<<N_MOLS, 64, 0, stream>>>(q, total_charge);

    // K3: per-molecule pairwise energy
    pair_energy<<<N_MOLS, 128, 0, stream>>>(xyz, q, energy);
}